// MyLoss_6330781795020
// MI455X (gfx1250) — compile-verified
//
#include <hip/hip_runtime.h>
#include <hip/hip_bf16.h>

#define BLOCK   256
#define NBLOCKS 2048
#define WAVES   (BLOCK / 32)

// native clang vector: required by __builtin_nontemporal_load,
// lowers to a single global_load_b128 with TH=NT
typedef float v4f __attribute__((ext_vector_type(4)));

struct BoxParams {
    float b0, b1, b2, b3;   // box
    float l0, l1;           // label box origin
    float sx, sy;           // label_w/ori_w, label_h/ori_h
};

__device__ __forceinline__ BoxParams load_params(const float* __restrict__ box,
                                                 const float* __restrict__ lbox) {
    BoxParams p;
    p.b0 = box[0];  p.b1 = box[1];  p.b2 = box[2];  p.b3 = box[3];
    p.l0 = lbox[0]; p.l1 = lbox[1];
    const float lw = lbox[2] - lbox[0];
    const float lh = lbox[3] - lbox[1];
    p.sx = lw / (p.b2 - p.b0);
    p.sy = lh / (p.b3 - p.b1);
    return p;
}

__device__ __forceinline__ void accum_point(float x0, float y0, float x1, float y1,
                                            const BoxParams& p,
                                            float& sum, float& cnt) {
    const bool inside = (p.b0 <= x0) & (x0 <= p.b2) & (p.b1 <= y0) & (y0 <= p.b3);
    const float nx = fmaf(x0 - p.b0, p.sx, p.l0);
    const float ny = fmaf(y0 - p.b1, p.sy, p.l1);
    const float dx = nx - x1;
    const float dy = ny - y1;
    const float d  = sqrtf(fmaf(dx, dx, dy * dy));
    sum += inside ? d : 0.0f;
    cnt += inside ? 1.0f : 0.0f;
}

__device__ __forceinline__ void accum_vec4(const v4f& a, const v4f& b,
                                           const BoxParams& p,
                                           float& sum, float& cnt) {
    // one v4f from each array = two (x,y) points
    accum_point(a.x, a.y, b.x, b.y, p, sum, cnt);
    accum_point(a.z, a.w, b.z, b.w, p, sum, cnt);
}

// Stage 1: streaming partial reduction. One (sum,cnt) pair per block -> d_ws.
__global__ void __launch_bounds__(BLOCK)
loss_partial_kernel(const v4f* __restrict__ p0, const v4f* __restrict__ p1,
                    const float* __restrict__ box, const float* __restrict__ lbox,
                    float* __restrict__ ws, int nv4) {
    const BoxParams p = load_params(box, lbox);

    float sum = 0.0f, cnt = 0.0f;
    const int stride = gridDim.x * BLOCK;
    int i = blockIdx.x * BLOCK + threadIdx.x;

    // 4-way unrolled grid-stride: 8 outstanding global_load_b128 per thread
    for (; i + 3 * stride < nv4; i += 4 * stride) {
        const v4f a0 = __builtin_nontemporal_load(&p0[i]);
        const v4f a1 = __builtin_nontemporal_load(&p0[i + stride]);
        const v4f a2 = __builtin_nontemporal_load(&p0[i + 2 * stride]);
        const v4f a3 = __builtin_nontemporal_load(&p0[i + 3 * stride]);
        const v4f c0 = __builtin_nontemporal_load(&p1[i]);
        const v4f c1 = __builtin_nontemporal_load(&p1[i + stride]);
        const v4f c2 = __builtin_nontemporal_load(&p1[i + 2 * stride]);
        const v4f c3 = __builtin_nontemporal_load(&p1[i + 3 * stride]);
        accum_vec4(a0, c0, p, sum, cnt);
        accum_vec4(a1, c1, p, sum, cnt);
        accum_vec4(a2, c2, p, sum, cnt);
        accum_vec4(a3, c3, p, sum, cnt);
    }
    for (; i < nv4; i += stride) {
        const v4f a = __builtin_nontemporal_load(&p0[i]);
        const v4f c = __builtin_nontemporal_load(&p1[i]);
        accum_vec4(a, c, p, sum, cnt);
    }

    // wave32 butterfly reduction (5 steps — gfx1250 is wave32, NOT wave64)
    for (int off = 16; off > 0; off >>= 1) {
        sum += __shfl_xor(sum, off, 32);
        cnt += __shfl_xor(cnt, off, 32);
    }

    __shared__ float ssum[WAVES];
    __shared__ float scnt[WAVES];
    const int wave = threadIdx.x >> 5;
    const int lane = threadIdx.x & 31;
    if (lane == 0) { ssum[wave] = sum; scnt[wave] = cnt; }
    __syncthreads();

    if (threadIdx.x == 0) {
        float s = 0.0f, c = 0.0f;
        #pragma unroll
        for (int w = 0; w < WAVES; ++w) { s += ssum[w]; c += scnt[w]; }
        ws[blockIdx.x]           = s;
        ws[NBLOCKS + blockIdx.x] = c;
    }
}

// Stage 2: single-block deterministic finalize (+ tail point if 2N % 4 == 2).
__global__ void __launch_bounds__(BLOCK)
loss_finalize_kernel(const float* __restrict__ ws,
                     const float* __restrict__ p0f, const float* __restrict__ p1f,
                     const float* __restrict__ box, const float* __restrict__ lbox,
                     float* __restrict__ out, int nfloats) {
    float s = 0.0f, c = 0.0f;
    for (int i = threadIdx.x; i < NBLOCKS; i += BLOCK) {
        s += ws[i];
        c += ws[NBLOCKS + i];
    }

    if (threadIdx.x == 0 && (nfloats & 3) == 2) {
        // one leftover point not covered by the vec4 stage
        const BoxParams p = load_params(box, lbox);
        const int base = nfloats - 2;
        accum_point(p0f[base], p0f[base + 1], p1f[base], p1f[base + 1], p, s, c);
    }

    __shared__ float ssum[BLOCK];
    __shared__ float scnt[BLOCK];
    ssum[threadIdx.x] = s;
    scnt[threadIdx.x] = c;
    __syncthreads();
    for (int half = BLOCK / 2; half > 0; half >>= 1) {
        if (threadIdx.x < half) {
            ssum[threadIdx.x] += ssum[threadIdx.x + half];
            scnt[threadIdx.x] += scnt[threadIdx.x + half];
        }
        __syncthreads();
    }

    if (threadIdx.x == 0) {
        const float sum = ssum[0];
        const float cnt = scnt[0];
        const float mean = sum / fmaxf(cnt, 1.0f);
        out[0] = (cnt > 0.0f) ? mean : 1000.0f;
    }
}

extern "C" void kernel_launch(void* const* d_in, const int* in_sizes, int n_in,
                              void* d_out, int out_size, void* d_ws, size_t ws_size,
                              hipStream_t stream) {
    const float* p0   = (const float*)d_in[0];   // pred0, (N,2) f32
    const float* p1   = (const float*)d_in[1];   // pred1, (N,2) f32
    const float* box  = (const float*)d_in[2];   // (1,4) f32
    const float* lbox = (const float*)d_in[3];   // (1,4) f32
    float* out = (float*)d_out;
    float* ws  = (float*)d_ws;                   // needs 2*NBLOCKS floats = 16 KB

    const int nfloats = in_sizes[0];             // 2*N
    const int nv4     = nfloats >> 2;            // vec4 count

    loss_partial_kernel<<<NBLOCKS, BLOCK, 0, stream>>>(
        (const v4f*)p0, (const v4f*)p1, box, lbox, ws, nv4);
    loss_finalize_kernel<<<1, BLOCK, 0, stream>>>(
        ws, p0, p1, box, lbox, out, nfloats);
}